// SIRENdecoder_decomp_16836271800900
// MI455X (gfx1250) — compile-verified
//
#include <hip/hip_runtime.h>

typedef __attribute__((ext_vector_type(2))) float v2f;
typedef __attribute__((ext_vector_type(4))) float v4f;
typedef __attribute__((ext_vector_type(8))) float v8f;

#define OMEGA0 30.0f
#define B_    8
#define N_    32768
#define C_    3
#define H_    256
#define R_    64
#define NHID_ 4
#define TILE  64
#define KTOT  (H_ + R_)   // 320: [activation | latent-scaled low-rank]

__device__ __forceinline__ v8f wmma4(v2f a, v2f b, v8f c) {
  // V_WMMA_F32_16X16X4_F32 : D = A(16x4) x B(4x16) + C, full fp32
  return __builtin_amdgcn_wmma_f32_16x16x4_f32(false, a, false, b, (short)0, c, false, false);
}

__global__ __launch_bounds__(256) void siren_fused(
    const float* __restrict__ coords, const float* __restrict__ latents,
    const float* __restrict__ Wf,     const float* __restrict__ bf,
    const float* __restrict__ Wh,     const float* __restrict__ bh,
    const float* __restrict__ Wl,     const float* __restrict__ bl,
    const float* __restrict__ U,      const float* __restrict__ V0,
    const float* __restrict__ Vh,     const float* __restrict__ HB,
    float* __restrict__ out)
{
  __shared__ float xs[TILE][KTOT];     // 64 x 320 fp32 = 80 KB (gfx1250: 320 KB/WGP)
  __shared__ float lat[R_];
  __shared__ float crd[TILE][C_];
  __shared__ float fbs[H_];            // per-layer FiLM bias + bias, cooperative

  const int tilesPerBatch = N_ / TILE;
  const int b  = blockIdx.x / tilesPerBatch;
  const int n0 = (blockIdx.x % tilesPerBatch) * TILE;

  const int tid  = threadIdx.x;
  const int wave = tid >> 5;
  const int lane = tid & 31;
  const int hi   = lane >> 4;   // half-wave select per WMMA layout
  const int ln   = lane & 15;

  // wave -> 2 row tiles x 4 col tiles of the 64x256 output (8 waves cover 4x16)
  const int m0a = (wave & 1) * 32;        // rows m0a..m0a+15
  const int m0b = m0a + 16;               // rows m0b..m0b+15
  const int ctg = wave >> 1;              // 0..3 -> col tiles ctg*4 .. ctg*4+3

  if (tid < R_)        lat[tid] = latents[b * R_ + tid];
  for (int idx = tid; idx < TILE * C_; idx += 256)
    ((float*)crd)[idx] = coords[(size_t)(b * N_ + n0) * C_ + idx];
  __syncthreads();

  // first layer's low-rank activation: t0 = (coords @ V0) * lat -> xs[:, 256:320]
  for (int idx = tid; idx < TILE * R_; idx += 256) {
    const int n = idx >> 6, r = idx & (R_ - 1);
    float s = crd[n][0] * V0[0 * R_ + r] + crd[n][1] * V0[1 * R_ + r]
            + crd[n][2] * V0[2 * R_ + r];
    xs[n][H_ + r] = s * lat[r];
  }

  // layer 0 = first layer (K=3 coords part via VALU, WMMA over K=256..319)
  // layers 1..4 = hidden (WMMA over full K=0..319)
  for (int l = 0; l <= NHID_; ++l) {
    const float* W   = (l == 0) ? Wf : (Wh + (size_t)(l - 1) * H_ * H_);
    const float* bb  = (l == 0) ? bf : (bh + (l - 1) * H_);
    const float* Ul  = U  + (size_t)l * H_ * R_;
    const float* HBl = HB + (size_t)l * H_ * R_;

    // cooperative FiLM bias: fbs[o] = b[o] + sum_r lat[r]*HB[o][r]
    // (one o per thread; contiguous HB row -> float4 loads; overlaps T-phase)
    {
      const float* hrow = HBl + (size_t)tid * R_;
      float fb = bb[tid];
      #pragma unroll
      for (int r = 0; r < R_; r += 4) {
        v4f h = *(const v4f*)&hrow[r];
        fb += lat[r] * h.x + lat[r + 1] * h.y + lat[r + 2] * h.z + lat[r + 3] * h.w;
      }
      fbs[tid] = fb;
    }

    if (l > 0) {
      // t = (xs[:,0:256] @ V_{l-1}) * lat -> xs[:,256:320]
      // 64x64 T = 4x4 tiles; wave owns 2 row tiles (m0a,m0b) x 1 col tile
      const float* Vl = Vh + (size_t)(l - 1) * H_ * R_;
      const int tct = ctg;                 // 0..3 col tiles of T
      v8f t0 = {}, t1 = {};
      for (int k = 0; k < H_; k += 4) {
        v2f a0 = *(const v2f*)&xs[m0a + ln][k + 2 * hi];
        v2f a1 = *(const v2f*)&xs[m0b + ln][k + 2 * hi];
        v2f bv;
        bv.x = Vl[(size_t)(k + 2 * hi)     * R_ + tct * 16 + ln];
        bv.y = Vl[(size_t)(k + 2 * hi + 1) * R_ + tct * 16 + ln];
        t0 = wmma4(a0, bv, t0);
        t1 = wmma4(a1, bv, t1);
      }
      // old xs[:,256:320] fully consumed before previous layer's store barrier;
      // T-phase reads only cols 0..255 -> safe to overwrite cols 256..319
      #pragma unroll
      for (int v = 0; v < 8; ++v) {
        const int r = tct * 16 + ln;
        xs[m0a + v + 8 * hi][H_ + r] = t0[v] * lat[r];
        xs[m0b + v + 8 * hi][H_ + r] = t1[v] * lat[r];
      }
    }
    __syncthreads();   // fbs (and T for l>0) visible to all waves

    // accumulator seed from LDS broadcast
    v8f acc0[4], acc1[4];
    #pragma unroll
    for (int j = 0; j < 4; ++j) {
      const float fb = fbs[(ctg * 4 + j) * 16 + ln];
      #pragma unroll
      for (int v = 0; v < 8; ++v) { acc0[j][v] = fb; acc1[j][v] = fb; }
    }

    if (l == 0) {
      // K=3 coords contribution via VALU (too thin for WMMA)
      #pragma unroll
      for (int j = 0; j < 4; ++j) {
        const int o = (ctg * 4 + j) * 16 + ln;
        const float w0 = Wf[o * 3 + 0], w1 = Wf[o * 3 + 1], w2 = Wf[o * 3 + 2];
        #pragma unroll
        for (int v = 0; v < 8; ++v) {
          const int ma = m0a + v + 8 * hi, mb = m0b + v + 8 * hi;
          acc0[j][v] += crd[ma][0] * w0 + crd[ma][1] * w1 + crd[ma][2] * w2;
          acc1[j][v] += crd[mb][0] * w0 + crd[mb][1] * w1 + crd[mb][2] * w2;
        }
      }
    } else {
      // dense part: K = 0..255 against W^T (B frag reused across both row tiles)
      for (int k = 0; k < H_; k += 4) {
        v2f a0 = *(const v2f*)&xs[m0a + ln][k + 2 * hi];
        v2f a1 = *(const v2f*)&xs[m0b + ln][k + 2 * hi];
        #pragma unroll
        for (int j = 0; j < 4; ++j) {
          const int o = (ctg * 4 + j) * 16 + ln;
          v2f bv = *(const v2f*)&W[(size_t)o * H_ + k + 2 * hi];
          acc0[j] = wmma4(a0, bv, acc0[j]);
          acc1[j] = wmma4(a1, bv, acc1[j]);
        }
      }
    }

    // low-rank part: K = 256..319 against U^T
    for (int k = 0; k < R_; k += 4) {
      v2f a0 = *(const v2f*)&xs[m0a + ln][H_ + k + 2 * hi];
      v2f a1 = *(const v2f*)&xs[m0b + ln][H_ + k + 2 * hi];
      #pragma unroll
      for (int j = 0; j < 4; ++j) {
        const int o = (ctg * 4 + j) * 16 + ln;
        v2f bv = *(const v2f*)&Ul[(size_t)o * R_ + k + 2 * hi];
        acc0[j] = wmma4(a0, bv, acc0[j]);
        acc1[j] = wmma4(a1, bv, acc1[j]);
      }
    }

    __syncthreads();   // all waves done reading xs
    #pragma unroll
    for (int j = 0; j < 4; ++j) {
      const int o = (ctg * 4 + j) * 16 + ln;
      #pragma unroll
      for (int v = 0; v < 8; ++v) {
        xs[m0a + v + 8 * hi][o] = __sinf(OMEGA0 * acc0[j][v]);
        xs[m0b + v + 8 * hi][o] = __sinf(OMEGA0 * acc1[j][v]);
      }
    }
    __syncthreads();
  }

  // final projection H -> 1 : 4 threads per point + in-wave shuffle reduce
  {
    const int p   = tid >> 2;       // point 0..63
    const int sub = tid & 3;        // 64-wide K slice per thread
    float a = 0.0f;
    #pragma unroll 4
    for (int i = sub * 64; i < sub * 64 + 64; ++i) a += xs[p][i] * Wl[i];
    a += __shfl_xor(a, 1);
    a += __shfl_xor(a, 2);
    if (sub == 0) out[(size_t)b * N_ + n0 + p] = a + bl[0];
  }
}

extern "C" void kernel_launch(void* const* d_in, const int* in_sizes, int n_in,
                              void* d_out, int out_size, void* d_ws, size_t ws_size,
                              hipStream_t stream) {
  const float* coords  = (const float*)d_in[0];
  const float* latents = (const float*)d_in[1];
  const float* Wf      = (const float*)d_in[2];
  const float* bf      = (const float*)d_in[3];
  const float* Wh      = (const float*)d_in[4];
  const float* bh      = (const float*)d_in[5];
  const float* Wl      = (const float*)d_in[6];
  const float* bl      = (const float*)d_in[7];
  const float* U       = (const float*)d_in[8];
  const float* V0      = (const float*)d_in[9];
  const float* Vh      = (const float*)d_in[10];
  const float* HB      = (const float*)d_in[11];
  float* out = (float*)d_out;

  dim3 grid(B_ * (N_ / TILE));
  siren_fused<<<grid, 256, 0, stream>>>(coords, latents, Wf, bf, Wh, bh,
                                        Wl, bl, U, V0, Vh, HB, out);
}